// InPlaneSplitLocallyConnected2D_12498354831988
// MI455X (gfx1250) — compile-verified
//
#include <hip/hip_runtime.h>
#include <hip/hip_bf16.h>
#include <stdint.h>

// Problem constants (from reference)
#define B_   32
#define H_   64
#define W_   64
#define C_   140
#define KH_  3
#define KW_  3
#define OH_  62
#define OW_  62
#define G_   2
#define GS_  70                 // channels per group
#define P_   (OH_ * OW_)        // 3844 output positions
#define KK_  9                  // 3x3 taps

// Tiling: one workgroup = one (b, oh, half-row). 62 = 2*31.
#define TW_    31                    // output columns per tile
#define IW_    (TW_ + KW_ - 1)       // 33 input columns needed
#define XSEG   (IW_ * C_)            // 4620 floats per input row-segment
#define ROWSTR (W_ * C_)             // 8960 floats between input rows
#define XTILE  (KH_ * XSEG)          // 13860 floats  (55,440 B)
#define WTILE  (G_ * TW_ * KK_)      // 558 floats    (2,232 B)

typedef float        v2f __attribute__((ext_vector_type(2)));
typedef unsigned int v4u __attribute__((ext_vector_type(4)));
typedef unsigned int v8u __attribute__((ext_vector_type(8)));

__global__ __launch_bounds__(256) void
lc2d_split_kernel(const float* __restrict__ x,
                  const float* __restrict__ kern,
                  float* __restrict__ out)
{
    __shared__ float smem[XTILE + WTILE];   // 57,672 bytes

    const int tid = threadIdx.x;
    int blk = blockIdx.x;
    const int t  = blk & 1;          // which half-row tile
    blk >>= 1;
    const int oh = blk % OH_;
    const int b  = blk / OH_;

    // LDS byte offsets (generic-pointer low 32 bits == workgroup LDS offset)
    const unsigned lds_x = (unsigned)(uintptr_t)(void*)smem;
    const unsigned lds_w = lds_x + (unsigned)(XTILE * 4);

    // ---- Phase 1a: one TDM descriptor moves the whole 3x4620-float x tile ----
    // D# group0: count=1, lds_addr, global tile address, type=2 ("image")
    // D# group1: data_size=4B, tensor_dim0=tile_dim0=4620, tensor_dim1=tile_dim1=3,
    //            tensor_dim0_stride=8960 floats (input row pitch). 2D -> groups 2/3 NULL.
    if (tid < 32) {                  // wave 0 only (uniform branch under wave32)
        const uint64_t ga = (uint64_t)(uintptr_t)
            (x + ((size_t)(b * H_ + oh) * W_ + (size_t)t * TW_) * C_);

        v4u g0;
        g0.x = 1u;                                   // count=1, user descriptor
        g0.y = lds_x;                                // lds_addr
        g0.z = (unsigned)ga;                         // global_addr[31:0]
        g0.w = (unsigned)(ga >> 32) | 0x80000000u;   // global_addr[56:32] | type=2<<30

        v8u g1;
        g1[0] = 2u << 16;                            // data_size = 4 bytes
        g1[1] = ((unsigned)XSEG & 0xFFFFu) << 16;    // tensor_dim0[15:0]
        g1[2] = ((unsigned)KH_ << 16) |              // tensor_dim1[15:0]
                ((unsigned)XSEG >> 16);              // tensor_dim0[31:16]
        g1[3] = ((unsigned)XSEG & 0xFFFFu) << 16;    // tile_dim0 (dim1 hi = 0)
        g1[4] = (unsigned)KH_;                       // tile_dim1=3, tile_dim2=0
        g1[5] = (unsigned)ROWSTR;                    // tensor_dim0_stride[31:0]
        g1[6] = 0u;                                  // stride hi / dim1_stride lo
        g1[7] = 0u;                                  // dim1_stride hi

        asm volatile("tensor_load_to_lds %0, %1"
                     :: "s"(g0), "s"(g1)
                     : "memory");
    }

    // ---- Phase 1b: async-stage this tile's weights (2 groups x 31 pos x 9) ----
    {
        const int p0 = oh * OW_ + t * TW_;             // first position of tile
        for (int i = tid; i < WTILE; i += 256) {
            const int g = i / (TW_ * KK_);
            const int j = i - g * (TW_ * KK_);
            const unsigned goff =
                (unsigned)(((unsigned)g * (unsigned)(P_ * KK_) +
                            (unsigned)p0 * KK_ + (unsigned)j) * 4u);
            const unsigned loff = lds_w + (unsigned)i * 4u;
            asm volatile("global_load_async_to_lds_b32 %0, %1, %2"
                         :: "v"(loff), "v"(goff), "s"(kern)
                         : "memory");
        }
    }

    // Drain this wave's async copies + TDM transfer, then barrier.
    asm volatile("s_wait_asynccnt 0" ::: "memory");
    __builtin_amdgcn_s_wait_tensorcnt(0);
    __syncthreads();

    // ---- Phase 2: 31 positions x 70 float2 channel-pairs per tile ----
    const float* sw = smem + XTILE;
    const size_t orow =
        ((size_t)(b * OH_ + oh) * OW_ + (size_t)t * TW_) * C_;

    for (int idx = tid; idx < TW_ * (C_ / 2); idx += 256) {
        const int owl = idx / (C_ / 2);          // 0..30
        const int c2  = idx - owl * (C_ / 2);    // 0..69
        const int c   = c2 * 2;                  // even channel index
        const int g   = (c2 >= (GS_ / 2)) ? 1 : 0;

        const float* wp = &sw[(g * TW_ + owl) * KK_];

        float ax = 0.f, ay = 0.f;
#pragma unroll
        for (int r = 0; r < KH_; ++r) {
#pragma unroll
            for (int s = 0; s < KW_; ++s) {
                const v2f xv =
                    *(const v2f*)&smem[(r * IW_ + owl + s) * C_ + c];
                const float w = wp[r * KW_ + s];
                ax = fmaf(xv.x, w, ax);
                ay = fmaf(xv.y, w, ay);
            }
        }
        v2f acc;
        acc.x = ax;
        acc.y = ay;
        // Output is never re-read: keep it out of L2's way.
        __builtin_nontemporal_store(acc,
            (v2f*)&out[orow + (size_t)owl * C_ + c]);
    }
}

extern "C" void kernel_launch(void* const* d_in, const int* in_sizes, int n_in,
                              void* d_out, int out_size, void* d_ws, size_t ws_size,
                              hipStream_t stream) {
    const float* x    = (const float*)d_in[0];   // (32,64,64,140) fp32
    const float* kern = (const float*)d_in[1];   // (2,3844,9) fp32
    float* out        = (float*)d_out;           // (32,62,62,140) fp32

    const int nblocks = B_ * OH_ * 2;            // 3968 workgroups
    lc2d_split_kernel<<<dim3(nblocks), dim3(256), 0, stream>>>(x, kern, out);
}